// NodePooling_2362232013315
// MI455X (gfx1250) — compile-verified
//
#include <hip/hip_runtime.h>

typedef __attribute__((ext_vector_type(2))) float v2f;
typedef __attribute__((ext_vector_type(8))) float v8f;

#define PCOLS 512   // P*D = 8*64 columns per node (flattened)

// ---------------------------------------------------------------------------
// Kernel 1: exclusive prefix sum of counts -> per-graph start node offsets.
// One block, Hillis-Steele scan in LDS over 1024-wide chunks with a carry.
// ---------------------------------------------------------------------------
__global__ void __launch_bounds__(1024)
seg_offsets_kernel(const int* __restrict__ counts, int n_graphs,
                   int* __restrict__ offsets) {
  __shared__ int buf[1024];
  __shared__ int carry;
  const int tid = threadIdx.x;
  if (tid == 0) carry = 0;
  __syncthreads();
  for (int base = 0; base < n_graphs; base += 1024) {
    const int g = base + tid;
    const int v = (g < n_graphs) ? counts[g] : 0;
    buf[tid] = v;
    __syncthreads();
    for (int off = 1; off < 1024; off <<= 1) {
      int t = (tid >= off) ? buf[tid - off] : 0;
      __syncthreads();
      buf[tid] += t;
      __syncthreads();
    }
    const int incl = buf[tid];
    const int excl = incl - v + carry;
    if (g < n_graphs) offsets[g] = excl;
    __syncthreads();
    if (tid == 1023) carry += buf[1023];   // chunk total
    __syncthreads();
  }
}

// ---------------------------------------------------------------------------
// Kernel 2: per-graph mean pooling via WMMA column-sums.
//   D = ones(16x4) * B(4x16) + C  =>  every row of D is the 4-row column sum.
// One block per graph, 8 waves, each wave owns 64 columns (4 tiles of 16).
// ---------------------------------------------------------------------------
__global__ void __launch_bounds__(256)
node_pool_wmma_kernel(const float* __restrict__ feat,
                      const int* __restrict__ counts,
                      const int* __restrict__ offsets,
                      float* __restrict__ out) {
  const int g    = blockIdx.x;
  const int tid  = threadIdx.x;
  const int wave = tid >> 5;          // 0..7
  const int lane = tid & 31;
  const int half = lane >> 4;         // which 16-lane half of the wave
  const int l16  = lane & 15;         // N = lane % 16 column within a tile
  const int colbase = wave << 6;      // 64 columns per wave

  const int start = offsets[g];
  const int count = counts[g];

  // lane-fixed base: column (colbase + l16) of node `start`
  const float* __restrict__ base =
      feat + (size_t)start * PCOLS + colbase + l16;

  v8f acc0 = {}, acc1 = {}, acc2 = {}, acc3 = {};
  v2f ones; ones[0] = 1.0f; ones[1] = 1.0f;   // A = all-ones 16x4

  const int full = count & ~3;
#pragma unroll 2
  for (int n0 = 0; n0 < full; n0 += 4) {
    // B VGPR slot (vgpr, half) holds one node's 16 columns of this tile.
    const float* p0 = base + (size_t)(n0 + half) * PCOLS;       // slot vgpr0
    const float* p1 = p0 + 2 * (size_t)PCOLS;                    // slot vgpr1
    v2f b0, b1, b2, b3;
    b0[0] = __builtin_nontemporal_load(p0 +  0);   // tile 0
    b1[0] = __builtin_nontemporal_load(p0 + 16);   // tile 1
    b2[0] = __builtin_nontemporal_load(p0 + 32);   // tile 2
    b3[0] = __builtin_nontemporal_load(p0 + 48);   // tile 3
    b0[1] = __builtin_nontemporal_load(p1 +  0);
    b1[1] = __builtin_nontemporal_load(p1 + 16);
    b2[1] = __builtin_nontemporal_load(p1 + 32);
    b3[1] = __builtin_nontemporal_load(p1 + 48);
    acc0 = __builtin_amdgcn_wmma_f32_16x16x4_f32(false, ones, false, b0,
                                                 (short)0, acc0, false, false);
    acc1 = __builtin_amdgcn_wmma_f32_16x16x4_f32(false, ones, false, b1,
                                                 (short)0, acc1, false, false);
    acc2 = __builtin_amdgcn_wmma_f32_16x16x4_f32(false, ones, false, b2,
                                                 (short)0, acc2, false, false);
    acc3 = __builtin_amdgcn_wmma_f32_16x16x4_f32(false, ones, false, b3,
                                                 (short)0, acc3, false, false);
  }

  // Remainder (count % 4): zero-fill missing rows; WMMA under full EXEC.
  if (full < count) {
    v2f b0 = {}, b1 = {}, b2 = {}, b3 = {};
    const int r0 = full + half;
    const int r1 = full + 2 + half;
    if (r0 < count) {
      const float* p0 = base + (size_t)r0 * PCOLS;
      b0[0] = p0[0]; b1[0] = p0[16]; b2[0] = p0[32]; b3[0] = p0[48];
    }
    if (r1 < count) {
      const float* p1 = base + (size_t)r1 * PCOLS;
      b0[1] = p1[0]; b1[1] = p1[16]; b2[1] = p1[32]; b3[1] = p1[48];
    }
    acc0 = __builtin_amdgcn_wmma_f32_16x16x4_f32(false, ones, false, b0,
                                                 (short)0, acc0, false, false);
    acc1 = __builtin_amdgcn_wmma_f32_16x16x4_f32(false, ones, false, b1,
                                                 (short)0, acc1, false, false);
    acc2 = __builtin_amdgcn_wmma_f32_16x16x4_f32(false, ones, false, b2,
                                                 (short)0, acc2, false, false);
    acc3 = __builtin_amdgcn_wmma_f32_16x16x4_f32(false, ones, false, b3,
                                                 (short)0, acc3, false, false);
  }

  // D layout: VGPR0 lanes 0-15 -> row M=0, N=lane; lanes 16-31 -> M=8,
  // N=lane-16. All rows are identical column sums, so acc[0] works for all.
  const float inv = 1.0f / (float)((count > 0) ? count : 1);
  const float sA  = (half ? acc2[0] : acc0[0]) * inv;   // tiles 0 / 2
  const float sB  = (half ? acc3[0] : acc1[0]) * inv;   // tiles 1 / 3
  const int   cA  = colbase + (half ? 32 : 0) + l16;
  const int   cB  = cA + 16;
  // input col c = p*64 + d  ->  output col d*8 + p
  out[(size_t)g * PCOLS + ((cA & 63) << 3) + (cA >> 6)] = sA;
  out[(size_t)g * PCOLS + ((cB & 63) << 3) + (cB >> 6)] = sB;
}

// ---------------------------------------------------------------------------
extern "C" void kernel_launch(void* const* d_in, const int* in_sizes, int n_in,
                              void* d_out, int out_size, void* d_ws,
                              size_t ws_size, hipStream_t stream) {
  const float* feat   = (const float*)d_in[0];
  const int*   counts = (const int*)d_in[1];
  const int n_graphs  = in_sizes[1];
  int* offsets        = (int*)d_ws;          // n_graphs ints of scratch
  float* out          = (float*)d_out;

  seg_offsets_kernel<<<1, 1024, 0, stream>>>(counts, n_graphs, offsets);
  node_pool_wmma_kernel<<<n_graphs, 256, 0, stream>>>(feat, counts, offsets,
                                                      out);
}